// AttnDecoder_44590350467702
// MI455X (gfx1250) — compile-verified
//
#include <hip/hip_runtime.h>
#include <hip/hip_bf16.h>

// ---------------------------------------------------------------------------
// AttnDecoder single step for MI455X (gfx1250, wave32).
//   H=1024, VOCAB=32000, L=128, B=64
// All GEMMs run through one LDS-tiled V_WMMA_F32_16X16X4_F32 kernel with a
// double-buffered async (GLOBAL_LOAD_ASYNC_TO_LDS_B128) staging pipeline.
// ---------------------------------------------------------------------------

#define KT        32     // K tile per LDS stage
#define LDSP      36     // LDS row pitch (floats): 144B rows -> 16B aligned, conflict-free
#define HDIM      1024
#define BDIM      64
#define LDIM      128
#define VOCAB_N   32000

typedef float v2f __attribute__((ext_vector_type(2)));
typedef float v4f __attribute__((ext_vector_type(4)));
typedef float v8f __attribute__((ext_vector_type(8)));
typedef int   v4i __attribute__((ext_vector_type(4)));

// ---- async global->LDS staging (CDNA5), guarded so absence keeps sync path --
#if __has_builtin(__builtin_amdgcn_global_load_async_to_lds_b128)
  #define HAVE_ASYNC_LDS 1
  using gv4i_p = __attribute__((address_space(1))) v4i *;   // prints as "int4 __device__*"
  using lv4i_p = __attribute__((address_space(3))) v4i *;   // prints as "int4 __shared__*"
  #if __has_builtin(__builtin_amdgcn_s_wait_asynccnt)
    #define WAIT_ASYNC() __builtin_amdgcn_s_wait_asynccnt(0)
  #else
    #define WAIT_ASYNC() asm volatile("s_wait_asynccnt 0x0" ::: "memory")
  #endif
#else
  #define HAVE_ASYNC_LDS 0
  #define WAIT_ASYNC() ((void)0)
#endif

// ---- f32 16x16x4 WMMA (native CDNA5 op). Fallback: shfl emulation ----------
__device__ inline v8f wmma_f32_16x16x4(v2f a, v2f b, v8f c) {
#if __has_builtin(__builtin_amdgcn_wmma_f32_16x16x4_f32)
  return __builtin_amdgcn_wmma_f32_16x16x4_f32(
      /*neg_a=*/false, a, /*neg_b=*/false, b,
      /*c_mod=*/(short)0, c, /*reuse_a=*/false, /*reuse_b=*/false);
#else
  const int lane = threadIdx.x & 31;
  const int half = lane >> 4, lm = lane & 15;
  #pragma unroll
  for (int j = 0; j < 8; ++j) {
    const int m = half * 8 + j, n = lm;
    float acc = c[j];
    #pragma unroll
    for (int k = 0; k < 4; ++k) {
      float av = __shfl(a[k & 1], ((k >> 1) << 4) + m, 32);
      float bv = __shfl(b[k & 1], ((k >> 1) << 4) + n, 32);
      acc += av * bv;
    }
    c[j] = acc;
  }
  return c;
#endif
}

// ---------------------------------------------------------------------------
// Generic GEMM: D[M,N] = A[M,K] * B[N,K]^T  (+bias[n])
// Workgroup tile: 64(M) x 128(N). 256 threads = 8 waves; wave w owns the
// 16-wide N-sub w across all four 16-row M-subtiles (4 accumulators).
// Double-buffered LDS staging: tile k+1 is DMA'd while tile k feeds WMMAs.
// MODE 0: store D (+bias).          NTB: non-temporal B loads / D stores
// MODE 1: attention epilogue: row r = l*B+b (flat encoder_outputs order);
//         partial = sum_n tanh(c + hw2b[b,n]) * v[n]  -> atomicAdd(scores[b,l])
// ---------------------------------------------------------------------------
template <int MODE, bool NTB>
__global__ __launch_bounds__(256) void gemm_wmma_f32(
    const float* __restrict__ A, int lda,
    const float* __restrict__ Bw, int ldb,
    const float* __restrict__ bias,
    float* __restrict__ D, int ldd, int K,
    const float* __restrict__ hw2b,   // [B, H]   (MODE 1)
    const float* __restrict__ vvec,   // [H]      (MODE 1)
    float* __restrict__ scores)       // [B, L]   (MODE 1)
{
  __shared__ __align__(16) float lA[2][64 * LDSP];
  __shared__ __align__(16) float lB[2][128 * LDSP];

  const int tid  = threadIdx.x;
  const int wave = tid >> 5;
  const int lane = tid & 31;
  const int half = lane >> 4;
  const int lm   = lane & 15;
  const int Mbase = blockIdx.x * 64;
  const int Nbase = blockIdx.y * 128;

  const int lr = tid >> 3;          // 0..31 : row-group for tile staging
  const int lc = (tid & 7) << 2;    // 0,4..28 : float4 column

  v8f c[4] = {};

  // Stage one K-tile (A: 64x32, B: 128x32) into LDS buffer `buf`.
  auto stage_tile = [&](int k0, int buf) {
#if HAVE_ASYNC_LDS
    #pragma unroll
    for (int p = 0; p < 2; ++p) {
      const int row = lr + p * 32;
      __builtin_amdgcn_global_load_async_to_lds_b128(
          (gv4i_p)(uintptr_t)(A + (size_t)(Mbase + row) * lda + k0 + lc),
          (lv4i_p)(uint32_t)(uintptr_t)&lA[buf][row * LDSP + lc], 0, 0);
    }
    #pragma unroll
    for (int p = 0; p < 4; ++p) {
      const int row = lr + p * 32;
      __builtin_amdgcn_global_load_async_to_lds_b128(
          (gv4i_p)(uintptr_t)(Bw + (size_t)(Nbase + row) * ldb + k0 + lc),
          (lv4i_p)(uint32_t)(uintptr_t)&lB[buf][row * LDSP + lc], 0, 0);
    }
#else
    #pragma unroll
    for (int p = 0; p < 2; ++p) {
      const int row = lr + p * 32;
      v4f val = *(const v4f*)(A + (size_t)(Mbase + row) * lda + k0 + lc);
      *(v4f*)(&lA[buf][row * LDSP + lc]) = val;
    }
    #pragma unroll
    for (int p = 0; p < 4; ++p) {
      const int row = lr + p * 32;
      const v4f* gp = (const v4f*)(Bw + (size_t)(Nbase + row) * ldb + k0 + lc);
      v4f val = NTB ? __builtin_nontemporal_load(gp) : *gp;
      *(v4f*)(&lB[buf][row * LDSP + lc]) = val;
    }
    if (k0 + KT < K)   // pull tile-after-next toward the caches
      __builtin_prefetch(Bw + (size_t)(Nbase + lr) * ldb + k0 + KT, 0, 3);
#endif
  };

  int buf = 0;
  stage_tile(0, 0);

  for (int k0 = 0; k0 < K; k0 += KT) {
    WAIT_ASYNC();            // this wave's DMA into `buf` is complete
    __syncthreads();         // everyone's is; previous reads of buf^1 done
    if (k0 + KT < K) stage_tile(k0 + KT, buf ^ 1);

    // ---- 8 x (1 B-frag, 4 A-frags, 4 WMMA) over K=32 ----
    #pragma unroll
    for (int kb = 0; kb < KT; kb += 4) {
      v2f bfrag = *(const v2f*)(&lB[buf][(wave * 16 + lm) * LDSP + kb + half * 2]);
      #pragma unroll
      for (int mt = 0; mt < 4; ++mt) {
        v2f afrag = *(const v2f*)(&lA[buf][(mt * 16 + lm) * LDSP + kb + half * 2]);
        c[mt] = wmma_f32_16x16x4(afrag, bfrag, c[mt]);
      }
    }
    buf ^= 1;
  }

  const int ncol = Nbase + wave * 16 + lm;

  if (MODE == 0) {
    const float bv = bias ? bias[ncol] : 0.0f;
    #pragma unroll
    for (int mt = 0; mt < 4; ++mt) {
      #pragma unroll
      for (int j = 0; j < 8; ++j) {
        const int row = Mbase + mt * 16 + half * 8 + j;
        float* dp = &D[(size_t)row * ldd + ncol];
        if (NTB) __builtin_nontemporal_store(c[mt][j] + bv, dp);
        else     *dp = c[mt][j] + bv;
      }
    }
  } else {
    const float vv = vvec[ncol];
    #pragma unroll
    for (int mt = 0; mt < 4; ++mt) {
      #pragma unroll
      for (int j = 0; j < 8; ++j) {
        const int row = Mbase + mt * 16 + half * 8 + j;   // row = l*B + b
        const int b = row & (BDIM - 1);
        const int l = row >> 6;
        float t = tanhf(c[mt][j] + hw2b[b * HDIM + ncol]) * vv;
        #pragma unroll
        for (int off = 8; off >= 1; off >>= 1) t += __shfl_xor(t, off, 16);
        if (lm == 0) atomicAdd(&scores[b * LDIM + l], t);
      }
    }
  }
}

// ---------------------------------------------------------------------------
__global__ void zero_kernel(float* __restrict__ p, int n) {
  int i = blockIdx.x * blockDim.x + threadIdx.x;
  if (i < n) p[i] = 0.0f;
}

// softmax over L per batch row + context vector weighted[b,h] = sum_l p*eo[l,b,h]
__global__ __launch_bounds__(128) void softmax_weighted(
    const float* __restrict__ scores, const float* __restrict__ eo,
    float* __restrict__ attn_out, float* __restrict__ weighted)
{
  __shared__ float p[LDIM];
  __shared__ float red[4];
  const int b = blockIdx.x;
  const int l = threadIdx.x;           // 0..127
  const int wv = l >> 5, ln = l & 31;

  float s = scores[b * LDIM + l];
  float m = s;
  #pragma unroll
  for (int off = 16; off >= 1; off >>= 1) m = fmaxf(m, __shfl_xor(m, off, 32));
  if (ln == 0) red[wv] = m;
  __syncthreads();
  m = fmaxf(fmaxf(red[0], red[1]), fmaxf(red[2], red[3]));
  __syncthreads();

  float e = __expf(s - m);
  float sum = e;
  #pragma unroll
  for (int off = 16; off >= 1; off >>= 1) sum += __shfl_xor(sum, off, 32);
  if (ln == 0) red[wv] = sum;
  __syncthreads();
  sum = red[0] + red[1] + red[2] + red[3];

  const float pv = e / sum;
  p[l] = pv;
  attn_out[b * LDIM + l] = pv;
  __syncthreads();

  float acc[8] = {0.f, 0.f, 0.f, 0.f, 0.f, 0.f, 0.f, 0.f};
  for (int ll = 0; ll < LDIM; ++ll) {
    const float pw = p[ll];
    const float* row = eo + ((size_t)ll * BDIM + b) * HDIM;
    #pragma unroll
    for (int i = 0; i < 8; ++i) acc[i] += pw * row[threadIdx.x + 128 * i];
  }
  #pragma unroll
  for (int i = 0; i < 8; ++i)
    weighted[b * HDIM + threadIdx.x + 128 * i] = acc[i];
}

// embedding gather + build x=[emb,weighted] and out_cat tail sections
__global__ void build_x(const int* __restrict__ ids, const float* __restrict__ emb,
                        const float* __restrict__ weighted,
                        float* __restrict__ x, float* __restrict__ out_cat)
{
  const int idx = blockIdx.x * blockDim.x + threadIdx.x;   // 65536
  const int b = idx >> 10, h = idx & (HDIM - 1);
  const float e = emb[(size_t)ids[b] * HDIM + h];
  const float w = weighted[idx];
  x[b * 2048 + h]          = e;
  x[b * 2048 + HDIM + h]   = w;
  out_cat[b * 3072 + HDIM + h]     = w;
  out_cat[b * 3072 + 2 * HDIM + h] = e;
}

// GRU cell pointwise (PyTorch gate order r,z,n)
__global__ void gru_pointwise(const float* __restrict__ gx, const float* __restrict__ gh,
                              const float* __restrict__ h0,
                              float* __restrict__ h_out, float* __restrict__ out_cat)
{
  const int idx = blockIdx.x * blockDim.x + threadIdx.x;   // 65536
  const int b = idx >> 10, h = idx & (HDIM - 1);
  const float xr = gx[b * 3072 + h], xz = gx[b * 3072 + HDIM + h], xn = gx[b * 3072 + 2 * HDIM + h];
  const float hr = gh[b * 3072 + h], hz = gh[b * 3072 + HDIM + h], hn = gh[b * 3072 + 2 * HDIM + h];
  const float r = 1.0f / (1.0f + __expf(-(xr + hr)));
  const float z = 1.0f / (1.0f + __expf(-(xz + hz)));
  const float n = tanhf(xn + r * hn);
  const float hv = h0[idx];
  const float out = (1.0f - z) * n + z * hv;
  h_out[idx] = out;
  out_cat[b * 3072 + h] = out;
}

// ---------------------------------------------------------------------------
extern "C" void kernel_launch(void* const* d_in, const int* in_sizes, int n_in,
                              void* d_out, int out_size, void* d_ws, size_t ws_size,
                              hipStream_t stream) {
  (void)in_sizes; (void)n_in; (void)out_size; (void)ws_size;
  const int*   ids    = (const int*)  d_in[0];
  const float* hidden = (const float*)d_in[1];   // [1,B,H] -> [64,1024]
  const float* eo     = (const float*)d_in[2];   // [L,B,H] flat
  const float* emb    = (const float*)d_in[3];
  const float* attnW  = (const float*)d_in[4];   // [H, 2H]
  const float* attnb  = (const float*)d_in[5];
  const float* v      = (const float*)d_in[6];
  const float* w_ih   = (const float*)d_in[7];   // [3H, 2H]
  const float* w_hh   = (const float*)d_in[8];   // [3H, H]
  const float* b_ih   = (const float*)d_in[9];
  const float* b_hh   = (const float*)d_in[10];
  const float* out_W  = (const float*)d_in[11];  // [V, 3H]
  const float* out_b  = (const float*)d_in[12];

  float* out = (float*)d_out;
  float* out_logits = out;                       // [64, 32000]
  float* out_h      = out + 2048000;             // [1, 64, 1024]
  float* out_attn   = out + 2113536;             // [64, 1, 128]

  float* ws       = (float*)d_ws;
  float* scores   = ws;               // 8192
  float* hw2b     = ws + 8192;        // 65536
  float* weighted = ws + 73728;       // 65536
  float* x        = ws + 139264;      // 131072
  float* gx       = ws + 270336;      // 196608
  float* gh       = ws + 466944;      // 196608
  float* out_cat  = ws + 663552;      // 196608

  // 0) zero the attention-score accumulator (atomicAdd target)
  zero_kernel<<<32, 256, 0, stream>>>(scores, BDIM * LDIM);

  // 1) hw2b[b,h] = h0 . W2[h,:] + attn_b[h]   (W2 = attn_W[:, H:], row stride 2H)
  gemm_wmma_f32<0, false><<<dim3(1, 8), 256, 0, stream>>>(
      hidden, HDIM, attnW + HDIM, 2 * HDIM, attnb, hw2b, HDIM, HDIM,
      nullptr, nullptr, nullptr);

  // 2) scores[b,l] = sum_h tanh(eo[l,b,:].W1[h,:] + hw2b[b,h]) * v[h]
  //    (fused epilogue; never materializes the 33.5MB energy tensor)
  gemm_wmma_f32<1, false><<<dim3(128, 8), 256, 0, stream>>>(
      eo, HDIM, attnW, 2 * HDIM, nullptr, nullptr, 0, HDIM,
      hw2b, v, scores);

  // 3) softmax over L + context vector
  softmax_weighted<<<BDIM, 128, 0, stream>>>(scores, eo, out_attn, weighted);

  // 4) embedding gather, x = [embedded, weighted], fill out_cat[:,H:]
  build_x<<<256, 256, 0, stream>>>(ids, emb, weighted, x, out_cat);

  // 5) gx = x @ w_ih^T + b_ih ; gh = h0 @ w_hh^T + b_hh
  gemm_wmma_f32<0, false><<<dim3(1, 24), 256, 0, stream>>>(
      x, 2 * HDIM, w_ih, 2 * HDIM, b_ih, gx, 3 * HDIM, 2 * HDIM,
      nullptr, nullptr, nullptr);
  gemm_wmma_f32<0, false><<<dim3(1, 24), 256, 0, stream>>>(
      hidden, HDIM, w_hh, HDIM, b_hh, gh, 3 * HDIM, HDIM,
      nullptr, nullptr, nullptr);

  // 6) GRU pointwise -> h_new (also out_cat[:, :H])
  gru_pointwise<<<256, 256, 0, stream>>>(gx, gh, hidden, out_h, out_cat);

  // 7) logits = out_cat @ out_W^T + out_b   (HBM-bound: 393MB of out_W,
  //    streamed once with non-temporal policy so L2 keeps the reused tensors)
  gemm_wmma_f32<0, true><<<dim3(1, 250), 256, 0, stream>>>(
      out_cat, 3 * HDIM, out_W, 3 * HDIM, out_b, out_logits, VOCAB_N, 3 * HDIM,
      nullptr, nullptr, nullptr);
}